// SE3_layer_12824772346456
// MI455X (gfx1250) — compile-verified
//
#include <hip/hip_runtime.h>
#include <math.h>

typedef float v4f __attribute__((ext_vector_type(4)));

#define BLOCK            256
#define ROWS_PER_THREAD  2
#define ROWS_PER_BLOCK   (BLOCK * ROWS_PER_THREAD)        // 512 rows
#define TILE_V4          (ROWS_PER_BLOCK * 6 / 4)         // 768 x 16B = 12288 B

// One SE(3) exp row -> 3x v4f output {R00 R01 R02 t0 | R10.. t1 | R20.. t2}.
// Reference's reversed omega indexing: r = v[0:3], wz = v[3], wy = v[4], wx = v[5].
__device__ __forceinline__ void se3_row(float r0, float r1, float r2,
                                        float wz, float wy, float wx,
                                        v4f& o0, v4f& o1, v4f& o2) {
  const float t2 = fmaf(wx, wx, fmaf(wy, wy, wz * wz));
  const float theta = sqrtf(t2);
  float s, c;
  sincosf(theta, &s, &c);

  const float sdt = (theta > 0.0f) ? (s / theta) : 0.0f;

  const float t3 = t2 * theta;
  const float t4 = t2 * t2;
  const float t6 = t4 * t2;
  const float t8 = t4 * t4;

  const float A_big = (1.0f - c) / t2;                    // selected only when t2 > 0.1
  const float A_tay = 0.5f + t4 * (1.0f / 720.0f) + t8 * (1.0f / 3628800.0f)
                    - t2 * (1.0f / 24.0f) - t6 * (1.0f / 40320.0f);
  const float A = (t2 > 0.1f) ? A_big : ((t2 < 1e-6f) ? 0.0f : A_tay);

  const float B_big = (theta - s) / t3;                   // selected only when t3 > 0.1
  const float B_tay = (1.0f / 6.0f) + t4 * (1.0f / 5040.0f) + t8 * (1.0f / 39916800.0f)
                    - t2 * (1.0f / 120.0f) - t6 * (1.0f / 362880.0f);
  const float B = (t3 > 0.1f) ? B_big : ((t2 < 1e-6f) ? 0.0f : B_tay);

  // skew(w)^2 = w w^T - t2*I
  const float xx = wx * wx, yy = wy * wy, zz = wz * wz;
  const float xy = wx * wy, xz = wx * wz, yz = wy * wz;
  const float d0 = xx - t2, d1 = yy - t2, d2 = zz - t2;

  // R = I + sdt*S + A*S^2
  const float R00 = fmaf(A, d0, 1.0f);
  const float R01 = fmaf(A, xy, -sdt * wz);
  const float R02 = fmaf(A, xz,  sdt * wy);
  const float R10 = fmaf(A, xy,  sdt * wz);
  const float R11 = fmaf(A, d1, 1.0f);
  const float R12 = fmaf(A, yz, -sdt * wx);
  const float R20 = fmaf(A, xz, -sdt * wy);
  const float R21 = fmaf(A, yz,  sdt * wx);
  const float R22 = fmaf(A, d2, 1.0f);

  // V = I + A*S + B*S^2
  const float V00 = fmaf(B, d0, 1.0f);
  const float V01 = fmaf(B, xy, -A * wz);
  const float V02 = fmaf(B, xz,  A * wy);
  const float V10 = fmaf(B, xy,  A * wz);
  const float V11 = fmaf(B, d1, 1.0f);
  const float V12 = fmaf(B, yz, -A * wx);
  const float V20 = fmaf(B, xz, -A * wy);
  const float V21 = fmaf(B, yz,  A * wx);
  const float V22 = fmaf(B, d2, 1.0f);

  // t = V * rho
  const float tr0 = fmaf(V00, r0, fmaf(V01, r1, V02 * r2));
  const float tr1 = fmaf(V10, r0, fmaf(V11, r1, V12 * r2));
  const float tr2 = fmaf(V20, r0, fmaf(V21, r1, V22 * r2));

  o0 = (v4f){R00, R01, R02, tr0};
  o1 = (v4f){R10, R11, R12, tr1};
  o2 = (v4f){R20, R21, R22, tr2};
}

__global__ __launch_bounds__(BLOCK) void se3_exp_kernel(
    const float* __restrict__ in, float* __restrict__ out, int n_rows) {
  __shared__ v4f smem[TILE_V4];

  const int t = threadIdx.x;
  const long long block_row0 = (long long)blockIdx.x * ROWS_PER_BLOCK;

  if (block_row0 + ROWS_PER_BLOCK <= (long long)n_rows) {
    // ---- Full tile: CDNA5 async global->LDS DMA (no VGPR destinations) ----
    // WG-relative LDS offset = low 32 bits of the flat shared pointer.
    const unsigned lds_base = (unsigned)(uintptr_t)(void*)smem;
    const char* gbase = (const char*)in + block_row0 * 6 * sizeof(float);
#pragma unroll
    for (int seg = 0; seg < 3; ++seg) {
      const unsigned off = (unsigned)(seg * BLOCK + t) * 16u;  // coalesced 16B/lane
      const unsigned lds_addr = lds_base + off;
      const void* gaddr = gbase + off;
      asm volatile("global_load_async_to_lds_b128 %0, %1, off"
                   :: "v"(lds_addr), "v"(gaddr)
                   : "memory");
    }
    // Wait for this wave's async transfers, then make tile visible block-wide.
    asm volatile("s_wait_asynccnt 0x0" ::: "memory");
    __syncthreads();

    const v4f a = smem[3 * t + 0];
    const v4f b = smem[3 * t + 1];
    const v4f c = smem[3 * t + 2];

    v4f o0, o1, o2, o3, o4, o5;
    // row 2g:   v0..v5 = a.x a.y a.z a.w b.x b.y
    se3_row(a.x, a.y, a.z, a.w, b.x, b.y, o0, o1, o2);
    // row 2g+1: v0..v5 = b.z b.w c.x c.y c.z c.w
    se3_row(b.z, b.w, c.x, c.y, c.z, c.w, o3, o4, o5);

    const long long g = (long long)blockIdx.x * BLOCK + t;   // pair index
    v4f* out4 = (v4f*)out + g * 6;
    __builtin_nontemporal_store(o0, out4 + 0);
    __builtin_nontemporal_store(o1, out4 + 1);
    __builtin_nontemporal_store(o2, out4 + 2);
    __builtin_nontemporal_store(o3, out4 + 3);
    __builtin_nontemporal_store(o4, out4 + 4);
    __builtin_nontemporal_store(o5, out4 + 5);
  } else {
    // ---- Tail block (never taken for even 512-multiple N): scalar per row ----
#pragma unroll
    for (int k = 0; k < ROWS_PER_THREAD; ++k) {
      const long long r = block_row0 + (long long)t * ROWS_PER_THREAD + k;
      if (r < n_rows) {
        const float* p = in + r * 6;
        v4f o0, o1, o2;
        se3_row(p[0], p[1], p[2], p[3], p[4], p[5], o0, o1, o2);
        v4f* out4 = (v4f*)(out + r * 12);                    // 48B rows: 16B aligned
        __builtin_nontemporal_store(o0, out4 + 0);
        __builtin_nontemporal_store(o1, out4 + 1);
        __builtin_nontemporal_store(o2, out4 + 2);
      }
    }
  }
}

extern "C" void kernel_launch(void* const* d_in, const int* in_sizes, int n_in,
                              void* d_out, int out_size, void* d_ws, size_t ws_size,
                              hipStream_t stream) {
  const float* vec = (const float*)d_in[0];
  float* out = (float*)d_out;
  const int n_rows = in_sizes[0] / 6;                        // 2,097,152
  const int grid = (n_rows + ROWS_PER_BLOCK - 1) / ROWS_PER_BLOCK;  // 4096 blocks
  se3_exp_kernel<<<grid, BLOCK, 0, stream>>>(vec, out, n_rows);
}